// FullNetwork_72035191488652
// MI455X (gfx1250) — compile-verified
//
#include <hip/hip_runtime.h>

namespace {

constexpr int kN  = 2000;
constexpr int kM1 = 100;
constexpr int kM2 = 5;
constexpr int kK1 = 20;
constexpr int kK2 = 20;
constexpr int kF1 = 5;
constexpr int kF2 = 25;
constexpr int kF3 = 45;
constexpr int kT  = 256;   // 8 wave32
constexpr int kW  = kT / 32;
constexpr float kR1 = 0.3f;
constexpr float kR2 = 1.0f;
constexpr float kR3 = 2.0f;

typedef float v2f __attribute__((ext_vector_type(2)));
typedef float v8f __attribute__((ext_vector_type(8)));

#if defined(__HIP_DEVICE_COMPILE__) && __has_builtin(__builtin_amdgcn_wmma_f32_16x16x4_f32)
#define USE_WMMA_F32 1
#else
#define USE_WMMA_F32 0
#endif

// ---- Farthest point sampling. Points + running min-distances live in registers
// (PP points per thread, fully unrolled). Per step: register scan -> wave32
// shuffle butterfly -> 8 per-wave partials in LDS -> uniform serial combine.
// Only 2 block barriers per step. First-index tie-break matches jnp.argmax. ----
template <int N, int PP>
__device__ void fps_select(const float* __restrict__ pts, int m,
                           int* __restrict__ out_idx,
                           float* __restrict__ s_rv, int* __restrict__ s_ri) {
  const int tid = threadIdx.x;
  const int lane = tid & 31;
  const int wave = tid >> 5;
  float px[PP], py[PP], pz[PP], mind[PP];
#pragma unroll
  for (int i = 0; i < PP; ++i) {
    const int p = tid + kT * i;
    if (p < N) {
      px[i] = pts[p * 3 + 0];
      py[i] = pts[p * 3 + 1];
      pz[i] = pts[p * 3 + 2];
      mind[i] = 1e10f;
    } else {
      px[i] = 0.f; py[i] = 0.f; pz[i] = 0.f;
      mind[i] = -3.0e38f;   // padding: never wins the argmax
    }
  }
  int last = 0;
  for (int it = 0; it < m; ++it) {
    if (tid == 0) out_idx[it] = last;
    const float lx = pts[last * 3 + 0];
    const float ly = pts[last * 3 + 1];
    const float lz = pts[last * 3 + 2];
    float bv = -3.0e38f;
    int bi = 0x7fffffff;
#pragma unroll
    for (int i = 0; i < PP; ++i) {
      const float dx = px[i] - lx, dy = py[i] - ly, dz = pz[i] - lz;
      const float d = dx * dx + dy * dy + dz * dz;
      const float mn = fminf(mind[i], d);
      mind[i] = mn;
      if (mn > bv) { bv = mn; bi = tid + kT * i; }  // strict >: earliest index kept
    }
#pragma unroll
    for (int s = 16; s >= 1; s >>= 1) {
      const float ov = __shfl_xor(bv, s, 32);
      const int oi = __shfl_xor(bi, s, 32);
      if (ov > bv || (ov == bv && oi < bi)) { bv = ov; bi = oi; }
    }
    if (lane == 0) { s_rv[wave] = bv; s_ri[wave] = bi; }
    __syncthreads();
    float wv = s_rv[0];
    int wi = s_ri[0];
#pragma unroll
    for (int w = 1; w < kW; ++w) {
      const float ov = s_rv[w];
      const int oi = s_ri[w];
      if (ov > wv || (ov == wv && oi < wi)) { wv = ov; wi = oi; }
    }
    last = wi;            // uniform across all threads, no broadcast needed
    __syncthreads();      // protect s_rv/s_ri reuse next step
  }
}

// ---- One wave per query. Distances cached once in registers (PP per lane,
// unrolled -> VGPR resident). k selection passes are mutation-free: pass k
// takes the min over candidates lex-greater than the previous (d, idx) winner. ----
template <int N, int PP>
__device__ void radius_knn_wave(const float* __restrict__ qpts, const float* __restrict__ pts,
                                int k, float r2, int q, int lane,
                                int* __restrict__ out_idx, unsigned char* __restrict__ out_mask) {
  const float qx = qpts[q * 3 + 0], qy = qpts[q * 3 + 1], qz = qpts[q * 3 + 2];
  float dc[PP];
#pragma unroll
  for (int i = 0; i < PP; ++i) {
    const int p = lane + 32 * i;
    float d = 3.0e38f;
    if (p < N) {
      const float dx = pts[p * 3 + 0] - qx;
      const float dy = pts[p * 3 + 1] - qy;
      const float dz = pts[p * 3 + 2] - qz;
      d = dx * dx + dy * dy + dz * dz;
    }
    dc[i] = d;
  }
  float pd = -1.0f;
  int pi = -1;
  for (int kk = 0; kk < k; ++kk) {
    float bv = 3.0e38f;
    int bi = 0x7fffffff;
#pragma unroll
    for (int i = 0; i < PP; ++i) {
      const int p = lane + 32 * i;
      const float d = dc[i];
      const bool after = (d > pd) || (d == pd && p > pi);
      if (after && d < bv) { bv = d; bi = p; }
    }
#pragma unroll
    for (int s = 16; s >= 1; s >>= 1) {
      const float ov = __shfl_xor(bv, s, 32);
      const int oi = __shfl_xor(bi, s, 32);
      if (ov < bv || (ov == bv && oi < bi)) { bv = ov; bi = oi; }
    }
    pd = bv;
    pi = bi;
    if (lane == 0) {
      out_idx[q * k + kk] = bi;
      out_mask[q * k + kk] = (bv <= r2) ? (unsigned char)1 : (unsigned char)0;
    }
  }
}

__global__ __launch_bounds__(kT) void pointnet_ae_kernel(
    const float* __restrict__ g_pts,
    const float* __restrict__ gW1, const float* __restrict__ gb1,
    const float* __restrict__ gW2, const float* __restrict__ gb2,
    const float* __restrict__ gW3, const float* __restrict__ gb3,
    const float* __restrict__ gD1, const float* __restrict__ gbD1,
    const float* __restrict__ gD2, const float* __restrict__ gbD2,
    const float* __restrict__ gD3, const float* __restrict__ gbD3,
    float* __restrict__ out) {
  __shared__ float s_pts[kN * 3];          // 24000 B
  __shared__ float s_chunk[2500];          // 10000 B : WMMA h1 staging / h2 staging
  __shared__ float s_rv[kW];
  __shared__ int s_ri[kW];
  __shared__ int s_idx1[kM1 * kK1];        // 8000 B
  __shared__ unsigned char s_m1[kM1 * kK1];
  __shared__ float s_samples[kM1 * 3];
  __shared__ int s_sidx[kM1];
  __shared__ float s_f1[kM1 * kF1];
  __shared__ int s_sidx2[kM2];
  __shared__ float s_s2[kM2 * 3];
  __shared__ int s_idx2[kM2 * kK2];
  __shared__ unsigned char s_m2[kM2 * kK2];
  __shared__ float s_f2[kM2 * kF2];
  __shared__ float s_h3[kM2 * kF3];
  __shared__ float s_lat[kF3];
  __shared__ float s_dec[kM2 * 3];
  __shared__ float s_cf[kM2 * kF2];
  __shared__ float s_dec2[kM1 * 3];
  __shared__ float s_cf2[kM1 * kF1];

  const int tid = threadIdx.x;
  const int lane = tid & 31;
  const int wave = tid >> 5;
  const int cloud = blockIdx.x;

  // ---- stage point cloud into LDS ----
  const float* gp = g_pts + (size_t)cloud * kN * 3;
  for (int t = tid; t < kN * 3; t += kT) s_pts[t] = gp[t];
  __syncthreads();

  // ---- FPS #1: 2000 -> 100 ----
  fps_select<kN, 8>(s_pts, kM1, s_sidx, s_rv, s_ri);
  for (int t = tid; t < kM1; t += kT) {
    const int s = s_sidx[t];
    s_samples[t * 3 + 0] = s_pts[s * 3 + 0];
    s_samples[t * 3 + 1] = s_pts[s * 3 + 1];
    s_samples[t * 3 + 2] = s_pts[s * 3 + 2];
  }
  __syncthreads();

  // ---- radius kNN #1: 100 queries vs 2000 pts, k=20 ----
  for (int q = wave; q < kM1; q += kW)
    radius_knn_wave<kN, 63>(s_samples, s_pts, kK1, kR1 * kR1, q, lane, s_idx1, s_m1);
  __syncthreads();

  // ---- FPS #2: 100 -> 5 ----
  fps_select<kM1, 1>(s_samples, kM2, s_sidx2, s_rv, s_ri);
  for (int t = tid; t < kM2; t += kT) {
    const int s = s_sidx2[t];
    s_s2[t * 3 + 0] = s_samples[s * 3 + 0];
    s_s2[t * 3 + 1] = s_samples[s * 3 + 1];
    s_s2[t * 3 + 2] = s_samples[s * 3 + 2];
  }
  __syncthreads();

  // ---- radius kNN #2: 5 queries vs 100 samples, k=20 ----
  if (wave < kM2)
    radius_knn_wave<kM1, 4>(s_s2, s_samples, kK2, kR2 * kR2, wave, lane, s_idx2, s_m2);
  __syncthreads();

  // ---- MLP1: rel(2000x3) @ W1(3x5) + b1, relu, masked max over groups of 20 ----
  const float invR1 = 1.0f / kR1;
  const int half = lane >> 4;
  const int nn = lane & 15;
#if USE_WMMA_F32
  // B fragment: W1 as K(4, zero-padded) x N(16, zero-padded) f32
  // VGPR0: lanes0-15 K=0, lanes16-31 K=2 ; VGPR1: lanes0-15 K=1, lanes16-31 K=3
  v2f bfrag;
  {
    float e0 = 0.f, e1 = 0.f;
    if (nn < kF1) {
      const int K0 = 2 * half;      // 0 or 2, always a real K
      const int K1i = 2 * half + 1; // 1 or 3 (3 = zero pad)
      e0 = gW1[K0 * kF1 + nn];
      e1 = (K1i < 3) ? gW1[K1i * kF1 + nn] : 0.f;
    }
    bfrag.x = e0;
    bfrag.y = e1;
  }
  const float biasv = (nn < kF1) ? gb1[nn] : 0.f;
#endif
  for (int ch = 0; ch < 5; ++ch) {        // 5 chunks of 20 samples = 400 rows = 25 tiles
    const int row0 = ch * 400;
#if USE_WMMA_F32
    for (int t = wave; t < 25; t += kW) {
      const int r = row0 + t * 16 + nn;   // A-fragment row for this lane
      const int s = r / kK1, j = r % kK1;
      const int pidx = s_idx1[s * kK1 + j];
      const float ax = (s_pts[pidx * 3 + 0] - s_samples[s * 3 + 0]) * invR1;
      const float ay = (s_pts[pidx * 3 + 1] - s_samples[s * 3 + 1]) * invR1;
      const float az = (s_pts[pidx * 3 + 2] - s_samples[s * 3 + 2]) * invR1;
      v2f afrag;
      afrag.x = half ? az : ax;           // lanes16-31: K=2 ; lanes0-15: K=0
      afrag.y = half ? 0.f : ay;          // lanes16-31: K=3(pad) ; lanes0-15: K=1
      v8f acc = {0.f, 0.f, 0.f, 0.f, 0.f, 0.f, 0.f, 0.f};
      acc = __builtin_amdgcn_wmma_f32_16x16x4_f32(false, afrag, false, bfrag,
                                                  (short)0, acc, false, false);
      if (nn < kF1) {
#pragma unroll
        for (int rv = 0; rv < 8; ++rv) {
          const int lr = t * 16 + rv + 8 * half;   // local row in chunk
          s_chunk[lr * kF1 + nn] = fmaxf(acc[rv] + biasv, 0.f);
        }
      }
    }
#else
    for (int t = tid; t < 400 * kF1; t += kT) {
      const int lr = t / kF1, c = t % kF1;
      const int r = row0 + lr;
      const int s = r / kK1, j = r % kK1;
      const int pidx = s_idx1[s * kK1 + j];
      float acc = gb1[c];
      for (int kk = 0; kk < 3; ++kk)
        acc += (s_pts[pidx * 3 + kk] - s_samples[s * 3 + kk]) * invR1 * gW1[kk * kF1 + c];
      s_chunk[lr * kF1 + c] = fmaxf(acc, 0.f);
    }
#endif
    __syncthreads();
    for (int t = tid; t < 20 * kF1; t += kT) {
      const int sl = t / kF1, c = t % kF1;
      const int s = ch * 20 + sl;
      float mx = -1e10f;
      for (int j = 0; j < kK1; ++j)
        if (s_m1[s * kK1 + j]) mx = fmaxf(mx, s_chunk[(sl * kK1 + j) * kF1 + c]);
      s_f1[s * kF1 + c] = mx;
    }
    __syncthreads();
  }

  // ---- MLP2: concat(rel2, f1[idx2]) (5x20x8) @ W2(8x25), relu, masked max ----
  for (int t = tid; t < kM2 * kK2 * kF2; t += kT) {   // 2500
    const int q = t / (kK2 * kF2);
    const int rem = t % (kK2 * kF2);
    const int j = rem / kF2, c = rem % kF2;
    const int nb = s_idx2[q * kK2 + j];
    float acc = gb2[c];
#pragma unroll
    for (int kk = 0; kk < 3; ++kk)
      acc += (s_samples[nb * 3 + kk] - s_s2[q * 3 + kk]) * gW2[kk * kF2 + c];  // /R2 = 1
#pragma unroll
    for (int kk = 0; kk < kF1; ++kk)
      acc += s_f1[nb * kF1 + kk] * gW2[(3 + kk) * kF2 + c];
    s_chunk[t] = fmaxf(acc, 0.f);
  }
  __syncthreads();
  for (int t = tid; t < kM2 * kF2; t += kT) {   // 125
    const int q = t / kF2, c = t % kF2;
    float mx = -1e10f;
    for (int j = 0; j < kK2; ++j)
      if (s_m2[q * kK2 + j]) mx = fmaxf(mx, s_chunk[(q * kK2 + j) * kF2 + c]);
    s_f2[t] = mx;
  }
  __syncthreads();

  // ---- MLP3 + global max -> latent(45) ----
  for (int t = tid; t < kM2 * kF3; t += kT) {   // 225
    const int q = t / kF3, c = t % kF3;
    float acc = gb3[c];
#pragma unroll
    for (int kk = 0; kk < 3; ++kk)
      acc += (s_s2[q * 3 + kk] * (1.0f / kR3)) * gW3[kk * kF3 + c];
    for (int kk = 0; kk < kF2; ++kk)
      acc += s_f2[q * kF2 + kk] * gW3[(3 + kk) * kF3 + c];
    s_h3[t] = fmaxf(acc, 0.f);
  }
  __syncthreads();
  for (int t = tid; t < kF3; t += kT) {
    float mx = -3.0e38f;
    for (int q = 0; q < kM2; ++q) mx = fmaxf(mx, s_h3[q * kF3 + t]);
    s_lat[t] = mx;
  }
  __syncthreads();

  // ---- Decoder D1: latent(45) @ D1(45x140) -> dec(5x3), cf(5x25) ----
  for (int t = tid; t < kM2 * (3 + kF2); t += kT) {   // 140
    float acc = gbD1[t];
    for (int kk = 0; kk < kF3; ++kk) acc += s_lat[kk] * gD1[kk * 140 + t];
    const int q = t / 28, c = t % 28;
    if (c < 3) s_dec[q * 3 + c] = acc;
    else s_cf[q * kF2 + (c - 3)] = fmaxf(acc, 0.f);
  }
  __syncthreads();

  // ---- Decoder D2: cf(5x25) @ D2(25x160) -> dec2(100x3), cf2(100x5) ----
  for (int t = tid; t < kM2 * 160; t += kT) {   // 800
    const int q = t / 160, u = t % 160;
    float acc = gbD2[u];
    for (int kk = 0; kk < kF2; ++kk) acc += s_cf[q * kF2 + kk] * gD2[kk * 160 + u];
    const int mrow = q * kK2 + u / 8;
    const int c = u % 8;
    if (c < 3) s_dec2[mrow * 3 + c] = acc;
    else s_cf2[mrow * kF1 + (c - 3)] = fmaxf(acc, 0.f);
  }
  __syncthreads();

  // ---- Decoder D3: cf2(100x5) @ D3(5x60) via WMMA, fused compose + store ----
  float* gout = out + (size_t)cloud * kN * 3;
#if USE_WMMA_F32
  for (int tp = wave; tp < 28; tp += kW) {   // 7 row tiles x 4 col blocks
    const int tr = tp >> 2;                  // row tile 0..6 (rows tr*16 .. +15)
    const int nblk = tp & 3;                 // col block, u0 = 16*nblk
    const int u = 16 * nblk + nn;            // output col for this lane
    const int r = tr * 16 + nn;              // A row for this lane
    const bool rok = (r < kM1);
    const bool uok = (u < 60);
    // A: K slices {0..3} and {4..7}; K>=5 zero-padded. Rows >=100 zero-padded.
    v2f a0, a1;
    a0.x = (rok ? (half ? s_cf2[r * kF1 + 2] : s_cf2[r * kF1 + 0]) : 0.f);
    a0.y = (rok ? (half ? s_cf2[r * kF1 + 3] : s_cf2[r * kF1 + 1]) : 0.f);
    a1.x = ((rok && !half) ? s_cf2[r * kF1 + 4] : 0.f);
    a1.y = 0.f;
    // B: D3 rows K, cols u. Cols >=60 zero-padded.
    v2f b0, b1;
    b0.x = (uok ? (half ? gD3[2 * 60 + u] : gD3[0 * 60 + u]) : 0.f);
    b0.y = (uok ? (half ? gD3[3 * 60 + u] : gD3[1 * 60 + u]) : 0.f);
    b1.x = ((uok && !half) ? gD3[4 * 60 + u] : 0.f);
    b1.y = 0.f;
    v8f acc = {0.f, 0.f, 0.f, 0.f, 0.f, 0.f, 0.f, 0.f};
    acc = __builtin_amdgcn_wmma_f32_16x16x4_f32(false, a0, false, b0, (short)0, acc, false, false);
    acc = __builtin_amdgcn_wmma_f32_16x16x4_f32(false, a1, false, b1, (short)0, acc, false, false);
    const float bias = uok ? gbD3[u] : 0.f;
    if (uok) {
      const int c = u % 3;
      const int pcol = u / 3;
#pragma unroll
      for (int rv = 0; rv < 8; ++rv) {
        const int mrow = tr * 16 + rv + 8 * half;
        if (mrow < kM1) {
          const float dec3 = acc[rv] + bias;
          const int p = mrow * kK1 + pcol;
          const int q = mrow / kK2;
          gout[p * 3 + c] =
              ((s_dec[q * 3 + c] * kR3 + s_dec2[mrow * 3 + c]) * kR2 + dec3) * kR1;
        }
      }
    }
  }
#else
  for (int t = tid; t < kM1 * 60; t += kT) {   // 6000
    const int mrow = t / 60, u = t % 60;
    float acc = gbD3[u];
#pragma unroll
    for (int kk = 0; kk < kF1; ++kk) acc += s_cf2[mrow * kF1 + kk] * gD3[kk * 60 + u];
    const int p = mrow * kK1 + u / 3;
    const int c = u % 3;
    const int q = mrow / kK2;
    gout[p * 3 + c] = ((s_dec[q * 3 + c] * kR3 + s_dec2[mrow * 3 + c]) * kR2 + acc) * kR1;
  }
#endif
}

}  // namespace

extern "C" void kernel_launch(void* const* d_in, const int* in_sizes, int n_in,
                              void* d_out, int out_size, void* d_ws, size_t ws_size,
                              hipStream_t stream) {
  (void)n_in; (void)out_size; (void)d_ws; (void)ws_size;
  const float* pts = (const float*)d_in[0];
  const float* W1 = (const float*)d_in[1];
  const float* b1 = (const float*)d_in[2];
  const float* W2 = (const float*)d_in[3];
  const float* b2 = (const float*)d_in[4];
  const float* W3 = (const float*)d_in[5];
  const float* b3 = (const float*)d_in[6];
  const float* D1 = (const float*)d_in[7];
  const float* bD1 = (const float*)d_in[8];
  const float* D2 = (const float*)d_in[9];
  const float* bD2 = (const float*)d_in[10];
  const float* D3 = (const float*)d_in[11];
  const float* bD3 = (const float*)d_in[12];
  float* out = (float*)d_out;
  const int nclouds = in_sizes[0] / (kN * 3);   // B = 256
  pointnet_ae_kernel<<<nclouds, kT, 0, stream>>>(pts, W1, b1, W2, b2, W3, b3,
                                                 D1, bD1, D2, bD2, D3, bD3, out);
}